// GrainGrowthSingleTimestep_35364760715416
// MI455X (gfx1250) — compile-verified
//
#include <hip/hip_runtime.h>
#include <stdint.h>

// Problem constants (reference: A=B=1, dx=dy=0.5 -> 1/dx^2 = 4, dt=0.005)
#define NBATCH 8
#define NGRAIN 20
#define NH 512
#define NW 512
#define ROWS 8            // rows per workgroup tile in pass 2
#define DTIME_C 0.005f
#define INV_D2 4.0f

typedef __attribute__((ext_vector_type(4))) float v4f;
typedef __attribute__((ext_vector_type(4))) int   v4i;

#define GLOBAL_AS __attribute__((address_space(1)))
#define LDS_AS    __attribute__((address_space(3)))

typedef GLOBAL_AS v4i* gptr_v4i;
typedef LDS_AS    v4i* lptr_v4i;

#if defined(__has_builtin)
#  if __has_builtin(__builtin_amdgcn_global_load_async_to_lds_b128)
#    define USE_ASYNC_BUILTIN 1
#  else
#    define USE_ASYNC_BUILTIN 0
#  endif
#  if __has_builtin(__builtin_amdgcn_s_wait_asynccnt)
#    define USE_WAIT_BUILTIN 1
#  else
#    define USE_WAIT_BUILTIN 0
#  endif
#else
#  define USE_ASYNC_BUILTIN 0
#  define USE_WAIT_BUILTIN 0
#endif

// 16B global -> LDS async copy (per lane). Tracked by ASYNCcnt.
__device__ __forceinline__ void async_copy16(const float* g, float* l) {
#if USE_ASYNC_BUILTIN
  __builtin_amdgcn_global_load_async_to_lds_b128(
      (gptr_v4i)(GLOBAL_AS void*)g,
      (lptr_v4i)(LDS_AS void*)l,
      /*offset=*/0, /*cpol=*/0);
#else
  uint32_t laddr = (uint32_t)(uintptr_t)(LDS_AS void*)l;  // LDS byte address
  uint64_t gaddr = (uint64_t)(uintptr_t)g;
  asm volatile("global_load_async_to_lds_b128 %0, %1, off"
               :: "v"(laddr), "v"(gaddr) : "memory");
#endif
}

__device__ __forceinline__ void wait_async_zero() {
#if USE_WAIT_BUILTIN
  __builtin_amdgcn_s_wait_asynccnt(0);
#else
  asm volatile("s_wait_asynccnt 0" ::: "memory");
#endif
}

// ---------------------------------------------------------------------------
// Pass 1: S(b,h,w) = sum_g eta^2.  One float4 per thread, loop over grains.
// Reads eta once (167.8 MB), writes S (8 MB) to workspace.
// ---------------------------------------------------------------------------
__global__ __launch_bounds__(256)
void gg_sumsq_kernel(const float* __restrict__ etas, float* __restrict__ sumsq) {
  const int i4 = blockIdx.x * blockDim.x + threadIdx.x;   // 0 .. B*H*W/4-1
  const int pix4_per_plane = NH * NW / 4;                 // 65536
  const int b  = i4 / pix4_per_plane;
  const int p4 = i4 - b * pix4_per_plane;
  const v4f* src = (const v4f*)etas + (size_t)b * NGRAIN * pix4_per_plane + p4;
  v4f acc = (v4f)0.0f;
#pragma unroll
  for (int g = 0; g < NGRAIN; ++g) {
    v4f e = src[(size_t)g * pix4_per_plane];
    acc += e * e;
  }
  ((v4f*)sumsq)[i4] = acc;
}

// ---------------------------------------------------------------------------
// Pass 2: one workgroup per (b, g, 8-row block).  Async-stage 10 halo'd eta
// rows + 8 sumsq rows into LDS, then compute the update and NT-store.
// ---------------------------------------------------------------------------
__global__ __launch_bounds__(256)
void gg_update_kernel(const float* __restrict__ etas,
                      const float* __restrict__ sumsq,
                      const float* __restrict__ Lp,
                      const float* __restrict__ Kp,
                      float* __restrict__ out) {
  __shared__ float s_eta[(ROWS + 2) * NW];  // 20 KB (rows r0-1 .. r0+ROWS, wrapped)
  __shared__ float s_sum[ROWS * NW];        // 16 KB

  const int tid = threadIdx.x;              // 0..255
  const int nrb = NH / ROWS;                // 64 row-blocks
  const int rb  = blockIdx.x % nrb;
  const int g   = (blockIdx.x / nrb) % NGRAIN;
  const int b   = blockIdx.x / (nrb * NGRAIN);
  const int r0  = rb * ROWS;
  const int plane = (b * NGRAIN + g) * NH * NW;   // < 2^26, fits int

  // --- async-stage eta tile with periodic halo: (ROWS+2) rows x 512 cols ---
  // 1280 float4 chunks, 256 lanes -> 5 issues; H-wrap folded into lane address.
#pragma unroll
  for (int it = 0; it < (ROWS + 2) * NW / 4 / 256; ++it) {
    const int chunk = it * 256 + tid;
    const int trow  = chunk >> 7;             // NW/4 = 128 chunks per row
    const int c0    = (chunk & 127) << 2;
    const int grow  = (r0 - 1 + trow + NH) & (NH - 1);
    async_copy16(etas + plane + grow * NW + c0, &s_eta[trow * NW + c0]);
  }
  // --- async-stage sumsq tile: ROWS rows x 512 cols (1024 chunks, 4 issues) ---
#pragma unroll
  for (int it = 0; it < ROWS * NW / 4 / 256; ++it) {
    const int chunk = it * 256 + tid;
    const int trow  = chunk >> 7;
    const int c0    = (chunk & 127) << 2;
    async_copy16(sumsq + (b * NH + r0 + trow) * NW + c0, &s_sum[trow * NW + c0]);
  }

  const float dtL = DTIME_C * fabsf(Lp[0]);
  const float ak  = fabsf(Kp[0]);

  wait_async_zero();   // this wave's async writes to LDS complete
  __syncthreads();     // all waves' data visible

  // --- compute: 4 iterations x 4 adjacent pixels per thread ---
#pragma unroll
  for (int it = 0; it < ROWS * NW / 4 / 256; ++it) {
    const int chunk = it * 256 + tid;
    const int trow  = chunk >> 7;
    const int c0    = (chunk & 127) << 2;
    v4f o;
#pragma unroll
    for (int j = 0; j < 4; ++j) {
      const int col = c0 + j;
      const int lc  = (col + NW - 1) & (NW - 1);   // periodic W
      const int rc  = (col + 1) & (NW - 1);
      const float c = s_eta[(trow + 1) * NW + col];
      const float l = s_eta[(trow + 1) * NW + lc];
      const float r = s_eta[(trow + 1) * NW + rc];
      const float u = s_eta[trow * NW + col];
      const float d = s_eta[(trow + 2) * NW + col];
      const float S = s_sum[trow * NW + col];
      const float lap    = INV_D2 * (l + r + u + d - 4.0f * c);
      const float dfdeta = c * c * c - c + 2.0f * c * (S - c * c);
      float v = c - dtL * (dfdeta - ak * lap);
      v = fminf(fmaxf(v, 0.0f), 1.0f);
      o[j] = v;
    }
    // Output is never re-read: non-temporal store keeps eta resident in L2.
    v4f* dst = (v4f*)(out + plane + (r0 + trow) * NW + c0);
    __builtin_nontemporal_store(o, dst);
  }
}

extern "C" void kernel_launch(void* const* d_in, const int* in_sizes, int n_in,
                              void* d_out, int out_size, void* d_ws, size_t ws_size,
                              hipStream_t stream) {
  const float* etas  = (const float*)d_in[0];  // [8,20,512,512] f32
  const float* L     = (const float*)d_in[1];  // [1]
  const float* kappa = (const float*)d_in[2];  // [1]
  float* out   = (float*)d_out;
  float* sumsq = (float*)d_ws;                 // needs B*H*W*4 = 8 MB scratch

  const int n4 = NBATCH * NH * NW / 4;         // 524288 float4s
  gg_sumsq_kernel<<<n4 / 256, 256, 0, stream>>>(etas, sumsq);

  const int blocks = NBATCH * NGRAIN * (NH / ROWS);  // 10240
  gg_update_kernel<<<blocks, 256, 0, stream>>>(etas, sumsq, L, kappa, out);
}